// TreeCRFLayer_77532749627713
// MI455X (gfx1250) — compile-verified
//
#include <hip/hip_runtime.h>
#include <math.h>

// TreeCRF belief propagation on a complete binary tree (depth 10, C=2047, L=4, B=512).
// Core op per edge: 4x4 log-domain mat-vec done exactly in exp-domain via
// V_WMMA_F32_16X16X4_F32:  D[yi, b] = sum_yj exp(pairs)[yi,yj] * exp(local-m)[b,yj]
//   A (16x4): rows 0-3 = exp(trans) matrix (rows 4-15 zeroed)
//   B (4x16): 16 batches of exp(local - max) vectors   -> all 16 output columns valid
//   D rows 0-3 -> msg_exp;  msg = log(D) + m

typedef __attribute__((ext_vector_type(2))) float v2f;
typedef __attribute__((ext_vector_type(8))) float v8f;

#define NC 2047   // nodes
#define NB 512    // batch
#define NTILE 32  // NB / 16

// per-batch max over the 4 labels + exp-domain locals.
// Lane l holds local[b = l&15, yj = 2*(l>>4) + {0,1}] (the B-operand layout).
__device__ __forceinline__ void prep_local(float lx, float ly,
                                           float& e0, float& e1, float& m) {
  float h = fmaxf(lx, ly);
  m = fmaxf(h, __shfl_xor(h, 16, 32));   // combine yj{0,1} with yj{2,3} half
  e0 = expf(lx - m);
  e1 = expf(ly - m);
}

// One tree edge for 16 batches: returns msg[yi=0..3] (valid in lanes 0..15, b=lane).
// trans matrix = pairs[row][col][:][:], 16 contiguous floats.
__device__ __forceinline__ float4 child_msg(const float* __restrict__ P,
                                            int row, int col,
                                            float e0, float e1, float m,
                                            int bb, int kk) {
  // A operand: lane l -> A[yi = l&15][yj = kk + v]; rows >=4 forced to 0
  const float* tp = P + ((size_t)row * NC + col) * 16 + ((bb & 3) << 2) + kk;
  float va = (bb < 4) ? 1.0f : 0.0f;
  v2f a; a.x = va * expf(tp[0]); a.y = va * expf(tp[1]);
  v2f b; b.x = e0; b.y = e1;
  v8f c = {0.f, 0.f, 0.f, 0.f, 0.f, 0.f, 0.f, 0.f};
  v8f d = __builtin_amdgcn_wmma_f32_16x16x4_f32(false, a, false, b,
                                                (short)0, c, false, false);
  float4 msg;
  msg.x = logf(d[0]) + m;
  msg.y = logf(d[1]) + m;
  msg.z = logf(d[2]) + m;
  msg.w = logf(d[3]) + m;
  return msg;
}

// out[base..base+3] = log_softmax(X + alpha + beta); alpha read from out scratch
// (internal nodes) or 0 (leaves, whose scratch was never written).
__device__ __forceinline__ void finalize_store(float* __restrict__ out,
                                               const float* __restrict__ X,
                                               size_t base, float4 beta, bool leaf) {
  float4 xv = *(const float4*)(X + base);
  float l0 = xv.x + beta.x, l1 = xv.y + beta.y;
  float l2 = xv.z + beta.z, l3 = xv.w + beta.w;
  if (!leaf) {
    float4 av = *(const float4*)(out + base);
    l0 += av.x; l1 += av.y; l2 += av.z; l3 += av.w;
  }
  float mm  = fmaxf(fmaxf(l0, l1), fmaxf(l2, l3));
  float lse = logf(expf(l0 - mm) + expf(l1 - mm) + expf(l2 - mm) + expf(l3 - mm)) + mm;
  float4 r; r.x = l0 - lse; r.y = l1 - lse; r.z = l2 - lse; r.w = l3 - lse;
  *(float4*)(out + base) = r;
}

// ---------------- Kernel 1: upward, levels 10 -> 4 (per 16-batch tile x subtree) ----
// LDS frontier: 96 slots of 64 floats (odd level region [0,64), even [64,96)) = 24 KB.
__global__ __launch_bounds__(256) void up_deep_k(const float* __restrict__ X,
                                                 const float* __restrict__ P,
                                                 float* __restrict__ alpha) {
  __shared__ float sm[96 * 64];
  const int b0 = blockIdx.x * 16;
  const int sub = blockIdx.y;                 // subtree rooted at node 7+sub (level 3)
  const int lane = threadIdx.x & 31;
  const int wv = threadIdx.x >> 5, nw = blockDim.x >> 5;
  const int bb = lane & 15, kk = (lane >> 4) << 1;

  for (int d = 10; d >= 4; --d) {
    const int npar = 1 << (d - 4);                       // parents in this subtree
    const int parBase = (1 << (d - 1)) - 1 + sub * npar; // global index of first parent
    const int rdBase = (d & 1) ? 0 : 64;                 // region of level d (children)
    const int wrBase = ((d - 1) & 1) ? 0 : 64;           // region of level d-1
    for (int jp = wv; jp < npar; jp += nw) {
      const int p = parBase + jp;
      float4 acc = {0.f, 0.f, 0.f, 0.f};
      for (int side = 0; side < 2; ++side) {
        const int c = 2 * p + 1 + side;
        const int jc = 2 * jp + side;
        const float* xp = X + (((size_t)(b0 + bb) * NC + c) << 2) + kk;
        float lx = xp[0], ly = xp[1];
        if (d < 10) {                                    // leaves have alpha = 0
          const float* sp = &sm[(rdBase + jc) * 64 + bb * 4 + kk];
          lx += sp[0]; ly += sp[1];
        }
        float e0, e1, m;
        prep_local(lx, ly, e0, e1, m);
        float4 msg = child_msg(P, p, c, e0, e1, m, bb, kk);
        acc.x += msg.x; acc.y += msg.y; acc.z += msg.z; acc.w += msg.w;
      }
      if (lane < 16) {
        if (d > 4) *(float4*)&sm[(wrBase + jp) * 64 + lane * 4] = acc;
        *(float4*)(alpha + (((size_t)(b0 + lane) * NC + p) << 2)) = acc;  // out scratch
      }
    }
    __syncthreads();
  }
}

// ---------------- Kernel 2: upward, levels 3 -> 1 (one wave per tile) ----------------
__global__ __launch_bounds__(32) void up_top_k(const float* __restrict__ X,
                                               const float* __restrict__ P,
                                               float* __restrict__ alpha) {
  const int b0 = blockIdx.x * 16;
  const int lane = threadIdx.x & 31;
  const int bb = lane & 15, kk = (lane >> 4) << 1;
  for (int d = 3; d >= 1; --d) {
    const int npar = 1 << (d - 1);
    for (int jp = 0; jp < npar; ++jp) {
      const int p = (1 << (d - 1)) - 1 + jp;
      float4 acc = {0.f, 0.f, 0.f, 0.f};
      for (int side = 0; side < 2; ++side) {
        const int c = 2 * p + 1 + side;
        const float* xp = X + (((size_t)(b0 + bb) * NC + c) << 2) + kk;
        const float* ap = alpha + (((size_t)(b0 + bb) * NC + c) << 2) + kk;
        float lx = xp[0] + ap[0], ly = xp[1] + ap[1];  // same-wave global RAW: in order
        float e0, e1, m;
        prep_local(lx, ly, e0, e1, m);
        float4 msg = child_msg(P, p, c, e0, e1, m, bb, kk);
        acc.x += msg.x; acc.y += msg.y; acc.z += msg.z; acc.w += msg.w;
      }
      if (lane < 16)
        *(float4*)(alpha + (((size_t)(b0 + lane) * NC + p) << 2)) = acc;
    }
  }
}

// ---------------- Kernel 3: downward levels 1..3 + finalize nodes 0..14 -------------
// Exports level-3 betas (8 nodes) to ws for the deep kernel. One wave per tile.
__global__ __launch_bounds__(32) void down_top_k(const float* __restrict__ X,
                                                 const float* __restrict__ P,
                                                 float* __restrict__ out,
                                                 float* __restrict__ bws) {
  __shared__ float sm[6 * 64];  // level-1 betas: slots 0-1, level-2: slots 2-5
  const int b0 = blockIdx.x * 16;
  const int lane = threadIdx.x & 31;
  const int bb = lane & 15, kk = (lane >> 4) << 1;

  if (lane < 16) {  // root: beta = 0
    float4 z = {0.f, 0.f, 0.f, 0.f};
    finalize_store(out, X, ((size_t)(b0 + lane) * NC) << 2, z, false);
  }
  for (int d = 1; d <= 3; ++d) {
    const int npar = 1 << (d - 1);
    for (int jp = 0; jp < npar; ++jp) {
      const int p = (1 << (d - 1)) - 1 + jp;
      const float* xp = X + (((size_t)(b0 + bb) * NC + p) << 2) + kk;
      float lx = xp[0], ly = xp[1];
      if (d > 1) {
        const int rs = (d == 2) ? jp : (2 + jp);
        lx += sm[rs * 64 + bb * 4 + kk];
        ly += sm[rs * 64 + bb * 4 + kk + 1];
      }
      float e0, e1, m;
      prep_local(lx, ly, e0, e1, m);
      for (int side = 0; side < 2; ++side) {
        const int c = 2 * p + 1 + side;
        float4 beta = child_msg(P, c, p, e0, e1, m, bb, kk);  // trans = pairs[c][p]
        if (lane < 16) {
          if (d < 3) {
            const int ws_ = (d == 1) ? (2 * jp + side) : (2 + 2 * jp + side);
            *(float4*)&sm[ws_ * 64 + lane * 4] = beta;
          } else {
            *(float4*)(bws + (((size_t)(c - 7) * NB + b0 + lane) << 2)) = beta;
          }
          finalize_store(out, X, ((size_t)(b0 + lane) * NC + c) << 2, beta, false);
        }
      }
    }
  }
}

// ---------------- Kernel 4: downward, levels 4 -> 10, finalize levels 4..10 ---------
__global__ __launch_bounds__(256) void down_deep_k(const float* __restrict__ X,
                                                   const float* __restrict__ P,
                                                   float* __restrict__ out,
                                                   const float* __restrict__ bws) {
  __shared__ float sm[96 * 64];
  const int b0 = blockIdx.x * 16;
  const int sub = blockIdx.y;
  const int lane = threadIdx.x & 31;
  const int wv = threadIdx.x >> 5, nw = blockDim.x >> 5;
  const int bb = lane & 15, kk = (lane >> 4) << 1;

  for (int d = 4; d <= 10; ++d) {
    const int npar = 1 << (d - 4);
    const int parBase = (1 << (d - 1)) - 1 + sub * npar;
    const int rdBase = ((d - 1) & 1) ? 0 : 64;  // region of level d-1 (parents)
    const int wrBase = (d & 1) ? 0 : 64;        // region of level d (children)
    for (int jp = wv; jp < npar; jp += nw) {
      const int p = parBase + jp;
      const float* xp = X + (((size_t)(b0 + bb) * NC + p) << 2) + kk;
      float lx = xp[0], ly = xp[1];
      if (d == 4) {  // level-3 parent betas come from ws (written by down_top_k)
        const float* wp = bws + (((size_t)sub * NB + b0 + bb) << 2) + kk;
        lx += wp[0]; ly += wp[1];
      } else {
        const float* sp = &sm[(rdBase + jp) * 64 + bb * 4 + kk];
        lx += sp[0]; ly += sp[1];
      }
      float e0, e1, m;
      prep_local(lx, ly, e0, e1, m);
      for (int side = 0; side < 2; ++side) {
        const int c = 2 * p + 1 + side;
        const int jc = 2 * jp + side;
        float4 beta = child_msg(P, c, p, e0, e1, m, bb, kk);
        if (lane < 16) {
          if (d < 10) *(float4*)&sm[(wrBase + jc) * 64 + lane * 4] = beta;
          finalize_store(out, X, ((size_t)(b0 + lane) * NC + c) << 2, beta, d == 10);
        }
      }
    }
    __syncthreads();
  }
}

extern "C" void kernel_launch(void* const* d_in, const int* in_sizes, int n_in,
                              void* d_out, int out_size, void* d_ws, size_t ws_size,
                              hipStream_t stream) {
  const float* X = (const float*)d_in[0];      // (512, 2047, 4) f32
  const float* P = (const float*)d_in[1];      // (2047, 2047, 4, 4) f32
  float* out = (float*)d_out;                  // (512, 2047, 4) f32; doubles as alpha scratch
  float* bws = (float*)d_ws;                   // level-3 betas: 8*512*4 floats = 64 KB

  dim3 gdeep(NTILE, 8);
  up_deep_k  <<<gdeep, 256, 0, stream>>>(X, P, out);
  up_top_k   <<<NTILE,  32, 0, stream>>>(X, P, out);
  down_top_k <<<NTILE,  32, 0, stream>>>(X, P, out, bws);
  down_deep_k<<<gdeep, 256, 0, stream>>>(X, P, out, bws);
}